// NCCLoss1_82265803587858
// MI455X (gfx1250) — compile-verified
//
#include <hip/hip_runtime.h>

typedef __attribute__((ext_vector_type(2))) float v2f;
typedef __attribute__((ext_vector_type(4))) float v4f;
typedef __attribute__((ext_vector_type(8))) float v8f;

#define WIN    9
#define HALO   4
#define TH     32           // output rows per workgroup
#define TW     128          // output cols per workgroup
#define IMH    512
#define IMW    512
#define BATCH  32
#define RROWS  (TH + 2*HALO)   // 40 halo rows
#define RCOLS  (TW + 2*HALO)   // 136 halo cols (multiple of 4)
#define HS     144             // padded stride of H rows (16B aligned, bank-split)
#define NQ     5
#define SJ_OFF (RROWS*RCOLS)          // 5440
#define H_OFF  (2*RROWS*RCOLS)        // 10880
#define SMEM_FLOATS (H_OFF + NQ*RROWS*HS)   // 10880 + 28800 = 39680 floats

// ---------------------------------------------------------------------------
// Kernel 1: per-tile box sums.
//   stage:      b128 global -> b128 LDS (zero-padded halo)
//   horizontal: 24-col window in registers (6x ds_load_b128 per array),
//               5 sliding 9-tap sums, b128 stores to LDS H[q]
//   vertical:   V(16x16) = A_band(16x24) * H(24x16) via
//               v_wmma_f32_16x16x4_f32, K accumulated in 6 chunks of 4
//   pointwise cc + block reduction of (1 - cc) into partial[].
// ---------------------------------------------------------------------------
__global__ __launch_bounds__(256) void ncc_tile_kernel(
    const float* __restrict__ I, const float* __restrict__ J,
    float* __restrict__ partial) {
  extern __shared__ float smem[];
  float* sI = smem;
  float* sJ = smem + SJ_OFF;

  const int tid = threadIdx.x;
  const int rowBase = blockIdx.y * TH;
  const int colBase = blockIdx.x * TW;
  const size_t imgOff = (size_t)blockIdx.z * (size_t)(IMH * IMW);

  // ---- stage zero-padded halo tiles of I and J into LDS (vec4) ------------
  // 40 rows x 34 groups of 4 floats = 1360 groups per array.
  for (int g = tid; g < RROWS * (RCOLS / 4); g += 256) {
    const int r  = g / (RCOLS / 4);
    const int c4 = (g - r * (RCOLS / 4)) * 4;
    const int gr  = rowBase + r - HALO;
    const int gc0 = colBase + c4 - HALO;
    v4f a = {}, b = {};
    if ((unsigned)gr < (unsigned)IMH) {
      if (gc0 >= 0 && gc0 + 3 < IMW) {
        const size_t gbase = imgOff + (size_t)gr * IMW + (size_t)gc0;
        a = *(const v4f*)(I + gbase);          // global_load_b128 (16B aligned)
        b = *(const v4f*)(J + gbase);
      } else {
#pragma unroll
        for (int e = 0; e < 4; ++e) {
          const int gc = gc0 + e;
          if ((unsigned)gc < (unsigned)IMW) {
            const size_t gidx = imgOff + (size_t)gr * IMW + (size_t)gc;
            a[e] = I[gidx];
            b[e] = J[gidx];
          }
        }
      }
    }
    *(v4f*)(sI + r * RCOLS + c4) = a;          // ds_store_b128
    *(v4f*)(sJ + r * RCOLS + c4) = b;
  }
  __syncthreads();

  // ---- horizontal 9-tap running sums of the 5 products into H[q] ----------
  // (row, 16-col chunk) pairs: 40 * 8 = 320; window = 24 halo cols.
  for (int p = tid; p < RROWS * (TW / 16); p += 256) {
    const int r  = p >> 3;        // TW/16 == 8
    const int c0 = (p & 7) << 4;
    const int base = r * RCOLS + c0;

    float a[24], b[24];
#pragma unroll
    for (int v = 0; v < 6; ++v) {
      const v4f ta = *(const v4f*)(sI + base + 4 * v);   // ds_load_b128
      const v4f tb = *(const v4f*)(sJ + base + 4 * v);
#pragma unroll
      for (int e = 0; e < 4; ++e) { a[4 * v + e] = ta[e]; b[4 * v + e] = tb[e]; }
    }

    float s0 = 0.f, s1 = 0.f, s2 = 0.f, s3 = 0.f, s4 = 0.f;
#pragma unroll
    for (int k = 0; k < WIN; ++k) {
      s0 += a[k]; s1 += b[k];
      s2 += a[k] * a[k]; s3 += b[k] * b[k]; s4 += a[k] * b[k];
    }

    const int hb = H_OFF + r * HS + c0;
#pragma unroll
    for (int g4 = 0; g4 < 4; ++g4) {           // 4 groups of 4 output cols
      v4f h0, h1, h2, h3, h4;
#pragma unroll
      for (int jj = 0; jj < 4; ++jj) {
        const int j = 4 * g4 + jj;
        h0[jj] = s0; h1[jj] = s1; h2[jj] = s2; h3[jj] = s3; h4[jj] = s4;
        if (j < 15) {
          const float a1 = a[j + WIN], b1 = b[j + WIN];
          const float a0 = a[j],       b0 = b[j];
          s0 += a1 - a0;
          s1 += b1 - b0;
          s2 += a1 * a1 - a0 * a0;
          s3 += b1 * b1 - b0 * b0;
          s4 += a1 * b1 - a0 * b0;
        }
      }
      *(v4f*)(smem + hb + 4 * g4 + 0 * RROWS * HS) = h0;   // ds_store_b128 x5
      *(v4f*)(smem + hb + 4 * g4 + 1 * RROWS * HS) = h1;
      *(v4f*)(smem + hb + 4 * g4 + 2 * RROWS * HS) = h2;
      *(v4f*)(smem + hb + 4 * g4 + 3 * RROWS * HS) = h3;
      *(v4f*)(smem + hb + 4 * g4 + 4 * RROWS * HS) = h4;
    }
  }
  __syncthreads();

  // ---- vertical 9-tap via WMMA: V = A_band(16x24) * H(24x16) --------------
  const int lane  = tid & 31;
  const int wv    = tid >> 5;          // wave id 0..7 -> tile column
  const int m     = lane & 15;         // A-matrix row / B-matrix col
  const int khalf = (lane >> 4) << 1;  // 0 for lanes 0-15, 2 for lanes 16-31
  const int col   = (wv << 4) + m;     // H column this lane reads
  const float inv81 = 1.0f / 81.0f;
  float lsum = 0.0f;

#pragma unroll
  for (int t = 0; t < 2; ++t) {        // two 16-row tiles per wave
    const int R0 = t << 4;             // H row base (needs rows R0 .. R0+23)
    v8f acc[NQ];
#pragma unroll
    for (int q = 0; q < NQ; ++q) acc[q] = (v8f){};

#pragma unroll
    for (int kc = 0; kc < 6; ++kc) {   // K = 24 in chunks of 4
      const int kb = (kc << 2) + khalf;
      // banded-ones A: A[m][r] = 1 iff r in [m, m+8]
      v2f a;
      a.x = (kb     >= m && kb     <= m + 8) ? 1.0f : 0.0f;
      a.y = (kb + 1 >= m && kb + 1 <= m + 8) ? 1.0f : 0.0f;
      const int h0 = H_OFF + (R0 + kb) * HS + col;

      // fetch all 5 quantities' (K, K+1) pairs first (ds_load_2addr_b32 x5),
      // then issue the 5-WMMA burst behind a single dscnt wait.
      v2f bq[NQ];
#pragma unroll
      for (int q = 0; q < NQ; ++q) {
        const int qo = q * RROWS * HS;
        bq[q].x = smem[h0 + qo];        // K = kb
        bq[q].y = smem[h0 + HS + qo];   // K = kb + 1
      }
#pragma unroll
      for (int q = 0; q < NQ; ++q) {
        acc[q] = __builtin_amdgcn_wmma_f32_16x16x4_f32(
            false, a, false, bq[q], (short)0, acc[q], false, false);
      }
    }

    // ---- pointwise cc on the 8 accumulator elements of this lane ----------
#pragma unroll
    for (int e = 0; e < 8; ++e) {
      const float Is = acc[0][e];
      const float Js = acc[1][e];
      const float I2 = acc[2][e];
      const float J2 = acc[3][e];
      const float IJ = acc[4][e];
      const float cross = IJ - Is * Js * inv81;
      const float iv    = I2 - Is * Is * inv81;
      const float jv    = J2 - Js * Js * inv81;
      const float cc    = cross * cross / (iv * jv + 1e-5f);
      lsum += 1.0f - cc;
    }
  }

  // ---- block reduction of lsum -> partial[block] --------------------------
  __syncthreads();                 // done reading H/sI/sJ; reuse smem
  smem[tid] = lsum;
  __syncthreads();
#pragma unroll
  for (int s = 128; s > 0; s >>= 1) {
    if (tid < s) smem[tid] += smem[tid + s];
    __syncthreads();
  }
  if (tid == 0) {
    const int bid = (blockIdx.z * gridDim.y + blockIdx.y) * gridDim.x + blockIdx.x;
    partial[bid] = smem[0];
  }
}

// ---------------------------------------------------------------------------
// Kernel 2: reduce per-block partials -> scalar loss = mean(1 - cc) * weight.
// ---------------------------------------------------------------------------
__global__ __launch_bounds__(256) void ncc_finalize_kernel(
    const float* __restrict__ partial, float* __restrict__ out, int n) {
  __shared__ float red[256];
  float s = 0.0f;
  for (int i = threadIdx.x; i < n; i += 256) s += partial[i];
  red[threadIdx.x] = s;
  __syncthreads();
#pragma unroll
  for (int st = 128; st > 0; st >>= 1) {
    if (threadIdx.x < st) red[threadIdx.x] += red[threadIdx.x + st];
    __syncthreads();
  }
  if (threadIdx.x == 0) {
    const float invN = 1.0f / ((float)BATCH * (float)IMH * (float)IMW);
    out[0] = red[0] * invN;   // LOSS_WEIGHT == 1.0
  }
}

extern "C" void kernel_launch(void* const* d_in, const int* in_sizes, int n_in,
                              void* d_out, int out_size, void* d_ws, size_t ws_size,
                              hipStream_t stream) {
  const float* I = (const float*)d_in[0];   // y_true
  const float* J = (const float*)d_in[1];   // y_pred
  float* out     = (float*)d_out;
  float* partial = (float*)d_ws;            // 2048 floats

  dim3 grid(IMW / TW, IMH / TH, BATCH);     // (4, 16, 32) = 2048 blocks
  const size_t smemBytes = (size_t)SMEM_FLOATS * sizeof(float);  // ~155 KB

  ncc_tile_kernel<<<grid, 256, smemBytes, stream>>>(I, J, partial);

  const int nblk = grid.x * grid.y * grid.z;
  ncc_finalize_kernel<<<1, 256, 0, stream>>>(partial, out, nblk);
}